// Encoder_35201551958350
// MI455X (gfx1250) — compile-verified
//
#include <hip/hip_runtime.h>
#include <math.h>

// ---------------------------------------------------------------------------
// Graph-net encoder (node MLP + edge MLP: Linear -> GELU -> Linear -> LN)
// fp32 end-to-end on CDNA5 V_WMMA_F32_16X16X4_F32.
// v2: each wave32 owns a 32-row x 128-col tile (2 M-subtiles x 8 N-tiles,
//     16 x v8f accumulators) so every B fragment (W1/W2 from L2) is reused
//     by two WMMAs -> halves L2 fragment bandwidth per output row.
// ---------------------------------------------------------------------------

typedef float v2f __attribute__((ext_vector_type(2)));
typedef float v8f __attribute__((ext_vector_type(8)));

#define HDIM     128
#define HSTRIDE  132   // even padded stride: 8B-aligned rows -> ds_load_2addr_b64
#define MROWS    32    // rows per wave

__device__ __forceinline__ float gelu_exact(float x) {
    return 0.5f * x * (1.0f + erff(x * 0.70710678118654752440f));
}

// Wave-local LDS producer->consumer fence (CDNA5 split DS counter).
__device__ __forceinline__ void wave_lds_fence() {
    asm volatile("s_wait_dscnt 0" ::: "memory");
}

// featLds: [32][FSTRIDE] normalized input rows (zero-padded K)
// hbuf   : [32][HSTRIDE] hidden-activation scratch
// stats  : [32][2] per-row {mu, rsigma}
template <int K1TILES, int K1, int FSTRIDE>
__device__ __forceinline__ void mlp_ln_tile32(
    float* featLds, float* hbuf, float* stats,
    const float* __restrict__ W1, const float* __restrict__ B1,
    const float* __restrict__ W2, const float* __restrict__ B2,
    const float* __restrict__ G,  const float* __restrict__ Bt,
    float* __restrict__ out, long long rowBase, long long rowsTotal, int lane)
{
    const int half   = lane >> 4;    // 0: K offsets {0,1}, 1: K offsets {2,3}
    const int klo    = half * 2;
    const int rowSel = lane & 15;    // A-matrix row within a 16-row subtile
    const int colIn  = lane & 15;    // B/C column within a 16-wide tile

    const v8f vzero = {};
    v8f acc[2][8];
#pragma unroll
    for (int m = 0; m < 2; ++m)
#pragma unroll
        for (int n = 0; n < 8; ++n) acc[m][n] = vzero;

    // ---- layer 1: [32 x K1] @ [K1 x 128] ----
#pragma unroll
    for (int kk = 0; kk < K1TILES; ++kk) {
        const int kb = kk * 4 + klo;
        v2f a0, a1;
        a0.x = featLds[rowSel * FSTRIDE + kb];
        a0.y = featLds[rowSel * FSTRIDE + kb + 1];
        a1.x = featLds[(rowSel + 16) * FSTRIDE + kb];
        a1.y = featLds[(rowSel + 16) * FSTRIDE + kb + 1];
#pragma unroll
        for (int n = 0; n < 8; ++n) {
            const int col = n * 16 + colIn;
            v2f b;
            b.x = (kb     < K1) ? W1[kb * HDIM + col]       : 0.0f;
            b.y = (kb + 1 < K1) ? W1[(kb + 1) * HDIM + col] : 0.0f;
            acc[0][n] = __builtin_amdgcn_wmma_f32_16x16x4_f32(
                false, a0, false, b, (short)0, acc[0][n], false, false);
            acc[1][n] = __builtin_amdgcn_wmma_f32_16x16x4_f32(
                false, a1, false, b, (short)0, acc[1][n], false, false);
        }
    }

    // ---- bias + exact GELU -> hbuf (layer-2 A matrix) ----
#pragma unroll
    for (int n = 0; n < 8; ++n) {
        const int col  = n * 16 + colIn;
        const float b1 = B1[col];
#pragma unroll
        for (int m = 0; m < 2; ++m) {
#pragma unroll
            for (int r = 0; r < 8; ++r) {
                const int row = m * 16 + r + half * 8;   // C layout: VGPR r -> M=r / r+8
                hbuf[row * HSTRIDE + col] = gelu_exact(acc[m][n][r] + b1);
            }
            acc[m][n] = vzero;
        }
    }
    wave_lds_fence();

    // ---- layer 2: [32 x 128] @ [128 x 128] (W2 stays L2-resident) ----
    for (int kk = 0; kk < HDIM / 4; ++kk) {
        const int kb = kk * 4 + klo;
        v2f a0, a1;
        a0.x = hbuf[rowSel * HSTRIDE + kb];
        a0.y = hbuf[rowSel * HSTRIDE + kb + 1];
        a1.x = hbuf[(rowSel + 16) * HSTRIDE + kb];
        a1.y = hbuf[(rowSel + 16) * HSTRIDE + kb + 1];
#pragma unroll
        for (int n = 0; n < 8; ++n) {
            const int col = n * 16 + colIn;
            v2f b;
            b.x = W2[kb * HDIM + col];
            b.y = W2[(kb + 1) * HDIM + col];
            acc[0][n] = __builtin_amdgcn_wmma_f32_16x16x4_f32(
                false, a0, false, b, (short)0, acc[0][n], false, false);
            acc[1][n] = __builtin_amdgcn_wmma_f32_16x16x4_f32(
                false, a1, false, b, (short)0, acc[1][n], false, false);
        }
    }

    // ---- + bias, spill tile to LDS for row-wise LayerNorm ----
#pragma unroll
    for (int n = 0; n < 8; ++n) {
        const int col  = n * 16 + colIn;
        const float b2 = B2[col];
#pragma unroll
        for (int m = 0; m < 2; ++m)
#pragma unroll
            for (int r = 0; r < 8; ++r) {
                const int row = m * 16 + r + half * 8;
                hbuf[row * HSTRIDE + col] = acc[m][n][r] + b2;
            }
    }
    wave_lds_fence();

    // ---- LayerNorm stats: each of the 32 lanes owns one row, two-pass ----
    {
        const float* rowp = hbuf + lane * HSTRIDE;
        float s = 0.0f;
        for (int c = 0; c < HDIM; ++c) s += rowp[c];
        const float mu = s * (1.0f / HDIM);
        float v = 0.0f;
        for (int c = 0; c < HDIM; ++c) { float d = rowp[c] - mu; v += d * d; }
        v *= (1.0f / HDIM);
        stats[lane * 2 + 0] = mu;
        stats[lane * 2 + 1] = rsqrtf(v + 1e-5f);
    }
    wave_lds_fence();

    // ---- scale/shift + coalesced store (32 lanes sweep the columns) ----
#pragma unroll
    for (int j = 0; j < 4; ++j) {
        const int col = j * 32 + lane;
        const float g  = G[col];
        const float bt = Bt[col];
#pragma unroll 4
        for (int r = 0; r < MROWS; ++r) {
            const long long row = rowBase + r;
            if (row < rowsTotal) {
                const float mu = stats[r * 2 + 0];
                const float rs = stats[r * 2 + 1];
                const float xv = hbuf[r * HSTRIDE + col];
                out[row * HDIM + col] = (xv - mu) * rs * g + bt;
            }
        }
    }
}

// ---------------------------------------------------------------------------
// Node encoder: feat = normalize(concat(x[20], pos[2], node_type[1], v[2]))
// ---------------------------------------------------------------------------
__global__ __launch_bounds__(64) void node_encoder_kernel(
    const float* __restrict__ x, const float* __restrict__ pos,
    const float* __restrict__ ntype, const float* __restrict__ vel,
    const float* __restrict__ mean, const float* __restrict__ stdv,
    const float* __restrict__ W1, const float* __restrict__ B1,
    const float* __restrict__ W2, const float* __restrict__ B2,
    const float* __restrict__ G,  const float* __restrict__ Bt,
    float* __restrict__ out, long long Nrows)
{
    constexpr int K1 = 25, K1T = 7, FST = 28;   // K padded 25 -> 28
    __shared__ float featLds[2][MROWS * FST];
    __shared__ float hbuf[2][MROWS * HSTRIDE];
    __shared__ float stats[2][2 * MROWS];

    const int lane = threadIdx.x & 31;
    const int wave = threadIdx.x >> 5;
    const long long tiles = (Nrows + (MROWS - 1)) / MROWS;
    const long long tile  = (long long)blockIdx.x * 2 + wave;
    if (tile >= tiles) return;                  // wave-uniform exit (EXEC stays all-1)
    const long long rowBase = tile * MROWS;

    {   // every lane builds one feature row
        const long long node = rowBase + lane;
        float f[FST];
#pragma unroll
        for (int d = 0; d < FST; ++d) f[d] = 0.0f;
        if (node < Nrows) {
#pragma unroll
            for (int d = 0; d < 20; ++d) f[d] = x[node * 20 + d];
            f[20] = pos[node * 2 + 0];
            f[21] = pos[node * 2 + 1];
            f[22] = ntype[node];
            f[23] = vel[node * 2 + 0];
            f[24] = vel[node * 2 + 1];
#pragma unroll
            for (int d = 0; d < K1; ++d) f[d] = (f[d] - mean[d]) / stdv[d];
        }
#pragma unroll
        for (int d = 0; d < FST; ++d) featLds[wave][lane * FST + d] = f[d];
    }
    wave_lds_fence();

    mlp_ln_tile32<K1T, K1, FST>(featLds[wave], hbuf[wave], stats[wave],
                                W1, B1, W2, B2, G, Bt, out, rowBase, Nrows, lane);
}

// ---------------------------------------------------------------------------
// Edge encoder: feat = normalize([rel, dist, 1/d, 1/d^2, 1/d^3, 1/d^4, rel/d])
// edge_index is jnp.int64 in the reference -> read as long long.
// ---------------------------------------------------------------------------
__global__ __launch_bounds__(64) void edge_encoder_kernel(
    const float* __restrict__ pos, const long long* __restrict__ eidx,
    const float* __restrict__ mean, const float* __restrict__ stdv,
    const float* __restrict__ W1, const float* __restrict__ B1,
    const float* __restrict__ W2, const float* __restrict__ B2,
    const float* __restrict__ G,  const float* __restrict__ Bt,
    float* __restrict__ out, long long Erows)
{
    constexpr int K1 = 9, K1T = 3, FST = 12;    // K padded 9 -> 12
    __shared__ float featLds[2][MROWS * FST];
    __shared__ float hbuf[2][MROWS * HSTRIDE];
    __shared__ float stats[2][2 * MROWS];

    const int lane = threadIdx.x & 31;
    const int wave = threadIdx.x >> 5;
    const long long tiles = (Erows + (MROWS - 1)) / MROWS;
    const long long tile  = (long long)blockIdx.x * 2 + wave;
    if (tile >= tiles) return;
    const long long rowBase = tile * MROWS;

    {   // every lane builds one edge's feature row
        const long long e = rowBase + lane;
        float f[FST];
#pragma unroll
        for (int d = 0; d < FST; ++d) f[d] = 0.0f;
        if (e < Erows) {
            const long long si = eidx[e];           // edge_index[0]
            const long long di = eidx[Erows + e];   // edge_index[1]
            const float rx = pos[di * 2 + 0] - pos[si * 2 + 0];
            const float ry = pos[di * 2 + 1] - pos[si * 2 + 1];
            const float dist = sqrtf(rx * rx + ry * ry);
            const float w1 = 1.0f / dist;
            const float w2 = w1 * w1;
            const float w3 = w2 * w1;
            const float w4 = w3 * w1;
            f[0] = rx;  f[1] = ry;  f[2] = dist;
            f[3] = w1;  f[4] = w2;  f[5] = w3;  f[6] = w4;
            f[7] = rx * w1;  f[8] = ry * w1;
#pragma unroll
            for (int d = 0; d < K1; ++d) f[d] = (f[d] - mean[d]) / stdv[d];
        }
#pragma unroll
        for (int d = 0; d < FST; ++d) featLds[wave][lane * FST + d] = f[d];
    }
    wave_lds_fence();

    mlp_ln_tile32<K1T, K1, FST>(featLds[wave], hbuf[wave], stats[wave],
                                W1, B1, W2, B2, G, Bt, out, rowBase, Erows, lane);
}

// ---------------------------------------------------------------------------
extern "C" void kernel_launch(void* const* d_in, const int* in_sizes, int n_in,
                              void* d_out, int out_size, void* d_ws, size_t ws_size,
                              hipStream_t stream)
{
    const float*     x     = (const float*)d_in[0];
    const float*     pos   = (const float*)d_in[1];
    const float*     ntype = (const float*)d_in[2];
    const float*     vel   = (const float*)d_in[3];
    const long long* eidx  = (const long long*)d_in[4];   // int64 in reference
    const float*     nmean = (const float*)d_in[5];
    const float*     nstd  = (const float*)d_in[6];
    const float*     emean = (const float*)d_in[7];
    const float*     estd  = (const float*)d_in[8];
    const float*     nW1   = (const float*)d_in[9];
    const float*     nb1   = (const float*)d_in[10];
    const float*     nW2   = (const float*)d_in[11];
    const float*     nb2   = (const float*)d_in[12];
    const float*     ng    = (const float*)d_in[13];
    const float*     nbeta = (const float*)d_in[14];
    const float*     eW1   = (const float*)d_in[15];
    const float*     eb1   = (const float*)d_in[16];
    const float*     eW2   = (const float*)d_in[17];
    const float*     eb2   = (const float*)d_in[18];
    const float*     eg    = (const float*)d_in[19];
    const float*     ebeta = (const float*)d_in[20];

    const long long N = in_sizes[2];          // node_type has N elements
    const long long E = in_sizes[4] / 2;      // edge_index is [2, E]

    float* outNode = (float*)d_out;
    float* outEdge = (float*)d_out + N * (long long)HDIM;

    const long long ntiles = (N + (MROWS - 1)) / MROWS;
    const long long etiles = (E + (MROWS - 1)) / MROWS;
    const dim3 blk(64);   // 2 wave32 per block, one 32-row tile per wave

    node_encoder_kernel<<<dim3((unsigned)((ntiles + 1) / 2)), blk, 0, stream>>>(
        x, pos, ntype, vel, nmean, nstd,
        nW1, nb1, nW2, nb2, ng, nbeta, outNode, N);

    edge_encoder_kernel<<<dim3((unsigned)((etiles + 1) / 2)), blk, 0, stream>>>(
        pos, eidx, emean, estd,
        eW1, eb1, eW2, eb2, eg, ebeta, outEdge, E);
}